// AttentionEdgeLayer_41850161332540
// MI455X (gfx1250) — compile-verified
//
#include <hip/hip_runtime.h>
#include <hip/hip_bf16.h>
#include <math.h>

// Problem constants (from reference): B=2, N=256, M=512, IN_F=256, OUT_F=128
#define BATCH 2
#define NN    256
#define MM    512
#define INF_  256
#define OUTF  128
#define NEG_BIG -1.0e12f

typedef __attribute__((ext_vector_type(2))) float v2f;
typedef __attribute__((ext_vector_type(8))) float v8f;

// ---------------------------------------------------------------------------
// Kernel 1: H = X @ W   (X viewed as [B*N, IN_F] = [512,256], W [256,128])
// fp32 WMMA 16x16x4. One 16x16 output tile per wave; 8 waves/block cover a
// 16-row strip across all 128 output columns. 32 blocks total.
// ---------------------------------------------------------------------------
__global__ __launch_bounds__(256) void k_gemm_XW(const float* __restrict__ X,
                                                 const float* __restrict__ W,
                                                 float* __restrict__ H) {
  const int lane = threadIdx.x & 31;
  const int wave = threadIdx.x >> 5;       // 0..7 -> column tile
  const int rowTile = blockIdx.x * 16;     // 0..496
  const int colTile = wave * 16;
  const int hi = lane >> 4;                // lane group select (K offset +2)
  const int r = rowTile + (lane & 15);     // A row this lane feeds
  const int c = colTile + (lane & 15);     // B/C column this lane feeds

  v8f acc0 = {};
  v8f acc1 = {};
  for (int k0 = 0; k0 < INF_; k0 += 8) {
    // fragment 0: K = k0..k0+3
    {
      const int ka = k0 + 2 * hi;
      v2f a = *(const v2f*)(X + (size_t)r * INF_ + ka);       // contiguous pair
      v2f b;
      b[0] = W[(size_t)ka * OUTF + c];
      b[1] = W[(size_t)(ka + 1) * OUTF + c];
      acc0 = __builtin_amdgcn_wmma_f32_16x16x4_f32(false, a, false, b,
                                                   (short)0, acc0, false, false);
    }
    // fragment 1: K = k0+4..k0+7 (independent accumulator -> overlap)
    {
      const int ka = k0 + 4 + 2 * hi;
      v2f a = *(const v2f*)(X + (size_t)r * INF_ + ka);
      v2f b;
      b[0] = W[(size_t)ka * OUTF + c];
      b[1] = W[(size_t)(ka + 1) * OUTF + c];
      acc1 = __builtin_amdgcn_wmma_f32_16x16x4_f32(false, a, false, b,
                                                   (short)0, acc1, false, false);
    }
  }
  v8f acc = acc0 + acc1;
  // C/D layout: VGPR j -> row (j + 8*hi), col = lane&15 within tile
#pragma unroll
  for (int j = 0; j < 8; ++j)
    H[(size_t)(rowTile + j + 8 * hi) * OUTF + c] = acc[j];
}

// ---------------------------------------------------------------------------
// Kernel 2: hdot[b,n] = H[b,n,:] . (a[0:128] + a[128:256])
// ---------------------------------------------------------------------------
__global__ __launch_bounds__(256) void k_hdot(const float* __restrict__ H,
                                              const float* __restrict__ avec,
                                              float* __restrict__ hdot) {
  const int row = blockIdx.x * blockDim.x + threadIdx.x;  // 0..511 (= b*N+n)
  const float* h = H + (size_t)row * OUTF;
  float s = 0.f;
#pragma unroll 4
  for (int f = 0; f < OUTF; ++f) s += h[f] * (avec[f] + avec[OUTF + f]);
  hdot[row] = s;
}

// ---------------------------------------------------------------------------
// Kernel 3: edot[b,m] = (sum_n A[b,n,m]*hdot[b,n]) / max(degree,1)
// (Em collapses to this mat-vec; only its dot with s is ever consumed.)
// ---------------------------------------------------------------------------
__global__ __launch_bounds__(256) void k_edot(const float* __restrict__ A,
                                              const float* __restrict__ hdot,
                                              float* __restrict__ edot) {
  const int idx = blockIdx.x * blockDim.x + threadIdx.x;  // 0..1023
  const int b = idx >> 9;
  const int m = idx & (MM - 1);
  const float* Ab = A + (size_t)b * NN * MM;
  const float* hb = hdot + b * NN;
  float deg = 0.f, num = 0.f;
  for (int n = 0; n < NN; ++n) {
    const float av = Ab[(size_t)n * MM + m];   // coalesced over m
    deg += av;
    num += av * hb[n];
  }
  edot[idx] = num / (deg == 0.f ? 1.f : deg);
}

// ---------------------------------------------------------------------------
// Kernel 4: masked softmax rows.
// e[b,n,m] = leaky( n<128 ? hdot[b, 2n+[m>=256]] : edot[b, 2(n-128)+[m>=256]] )
// attn = softmax_m( A>0 ? e : -1e12 ).  One wave (32 lanes) per row, 16 elems
// per lane, wave32 shfl_xor reductions.
// ---------------------------------------------------------------------------
__global__ __launch_bounds__(256) void k_attn(const float* __restrict__ A,
                                              const float* __restrict__ hdot,
                                              const float* __restrict__ edot,
                                              float* __restrict__ attn) {
  const int lane = threadIdx.x & 31;
  const int wave = threadIdx.x >> 5;
  const int row = blockIdx.x * 8 + wave;  // 0..511
  const int b = row >> 8;
  const int n = row & (NN - 1);

  float vL, vR;
  if (n < 128) {
    vL = hdot[b * NN + 2 * n];
    vR = hdot[b * NN + 2 * n + 1];
  } else {
    vL = edot[b * MM + 2 * (n - 128)];
    vR = edot[b * MM + 2 * (n - 128) + 1];
  }
  const float eL = vL > 0.f ? vL : 0.01f * vL;
  const float eR = vR > 0.f ? vR : 0.01f * vR;

  const float* Arow = A + (size_t)b * NN * MM + (size_t)n * MM;
  float ev[16];
  float mx = -3.0e38f;
#pragma unroll
  for (int i = 0; i < 16; ++i) {
    const int m = lane + 32 * i;
    const float e = (i < 8) ? eL : eR;      // i<8  <=>  m<256
    const float t = (Arow[m] > 0.f) ? e : NEG_BIG;
    ev[i] = t;
    mx = fmaxf(mx, t);
  }
  for (int off = 16; off; off >>= 1) mx = fmaxf(mx, __shfl_xor(mx, off, 32));
  float sum = 0.f;
#pragma unroll
  for (int i = 0; i < 16; ++i) {
    ev[i] = expf(ev[i] - mx);
    sum += ev[i];
  }
  for (int off = 16; off; off >>= 1) sum += __shfl_xor(sum, off, 32);
  const float inv = 1.0f / sum;
  float* Orow = attn + (size_t)b * NN * MM + (size_t)n * MM;
#pragma unroll
  for (int i = 0; i < 16; ++i) Orow[lane + 32 * i] = ev[i] * inv;
}

// ---------------------------------------------------------------------------
// Kernel 5: h_prime[b,m,f] = sum_n attn[b,n,m] * H[b,n,f]
// = attn_b^T [512x256] @ H_b [256x128] per batch, fp32 WMMA.
// grid.x = B * 32 row-tiles; 8 waves/block cover 128 output columns.
// ---------------------------------------------------------------------------
__global__ __launch_bounds__(256) void k_out(const float* __restrict__ attn,
                                             const float* __restrict__ H,
                                             float* __restrict__ out) {
  const int bt = blockIdx.x;               // 0..63
  const int b = bt >> 5;
  const int mTile = (bt & 31) * 16;
  const float* Ab = attn + (size_t)b * NN * MM;
  const float* Hb = H + (size_t)b * NN * OUTF;
  float* Ob = out + (size_t)b * MM * OUTF;

  const int lane = threadIdx.x & 31;
  const int wave = threadIdx.x >> 5;
  const int fTile = wave * 16;
  const int hi = lane >> 4;
  const int m = mTile + (lane & 15);
  const int f = fTile + (lane & 15);

  v8f acc0 = {};
  v8f acc1 = {};
  for (int k0 = 0; k0 < NN; k0 += 8) {
    {
      const int ka = k0 + 2 * hi;
      v2f a;  // A-frag element (row=m, K=n) = attn[n*MM + m]
      a[0] = Ab[(size_t)ka * MM + m];
      a[1] = Ab[(size_t)(ka + 1) * MM + m];
      v2f bv;
      bv[0] = Hb[(size_t)ka * OUTF + f];
      bv[1] = Hb[(size_t)(ka + 1) * OUTF + f];
      acc0 = __builtin_amdgcn_wmma_f32_16x16x4_f32(false, a, false, bv,
                                                   (short)0, acc0, false, false);
    }
    {
      const int ka = k0 + 4 + 2 * hi;
      v2f a;
      a[0] = Ab[(size_t)ka * MM + m];
      a[1] = Ab[(size_t)(ka + 1) * MM + m];
      v2f bv;
      bv[0] = Hb[(size_t)ka * OUTF + f];
      bv[1] = Hb[(size_t)(ka + 1) * OUTF + f];
      acc1 = __builtin_amdgcn_wmma_f32_16x16x4_f32(false, a, false, bv,
                                                   (short)0, acc1, false, false);
    }
  }
  v8f acc = acc0 + acc1;
#pragma unroll
  for (int j = 0; j < 8; ++j)
    Ob[(size_t)(mTile + j + 8 * hi) * OUTF + f] = acc[j];
}

// ---------------------------------------------------------------------------
// Launch. Workspace layout (floats):
//   H    : [512,128]  @ 0          (65536)
//   hdot : [512]      @ 65536      (512)
//   edot : [2,512]    @ 66048      (1024)
//   attn : [2,256,512]@ 67072      (262144)   total ~1.26 MB
// ---------------------------------------------------------------------------
extern "C" void kernel_launch(void* const* d_in, const int* in_sizes, int n_in,
                              void* d_out, int out_size, void* d_ws, size_t ws_size,
                              hipStream_t stream) {
  (void)in_sizes; (void)n_in; (void)out_size; (void)ws_size;
  const float* X = (const float*)d_in[0];   // [2,256,256]
  const float* A = (const float*)d_in[1];   // [2,256,512]
  const float* W = (const float*)d_in[2];   // [256,128]
  const float* a = (const float*)d_in[3];   // [256,1]
  float* ws   = (float*)d_ws;
  float* H    = ws;
  float* hdot = ws + 65536;
  float* edot = ws + 66048;
  float* attn = ws + 67072;
  float* out  = (float*)d_out;              // [2,512,128]

  k_gemm_XW<<<32, 256, 0, stream>>>(X, W, H);
  k_hdot  <<<2,  256, 0, stream>>>(H, a, hdot);
  k_edot  <<<4,  256, 0, stream>>>(A, hdot, edot);
  k_attn  <<<64, 256, 0, stream>>>(A, hdot, edot, attn);
  k_out   <<<64, 256, 0, stream>>>(attn, H, out);
}